// FFD_27874337751114
// MI455X (gfx1250) — compile-verified
//
#include <hip/hip_runtime.h>
#include <stdint.h>

// Problem constants (match reference)
#define N_PTS     200000
#define N_CTRL    4096
#define NNZ_TOT   12800000
#define XS_ELEMS  (2 * N_CTRL * 3)     // 24576 floats = 96 KB (one LDS alloc)
#define XS_B1     (N_CTRL * 3)         // 12288: batch-1 offset inside LDS table
#define OUT_ELEMS (2 * N_PTS * 3)      // 1200000 floats
#define OUT_B1    (N_PTS * 3)          // 600000: batch-1 offset in output

// Tiling: 20 consecutive nnz per thread -> 5120 nnz per 256-thread block-tile.
// 12.8M / 5120 = 2500 tiles exactly; 500 persistent blocks x exactly 5 tiles.
#define TPB       256
#define IPT       20
#define TILE_NNZ  (TPB * IPT)           // 5120
#define NTILES    (NNZ_TOT / TILE_NNZ)  // 2500 (exact)
#define NBLOCKS   500

typedef __attribute__((ext_vector_type(4))) int          i32x4v;
typedef __attribute__((ext_vector_type(8))) int          i32x8v;
typedef __attribute__((ext_vector_type(4))) unsigned int u32x4v;
typedef __attribute__((ext_vector_type(4))) float        f32x4v;

__device__ __forceinline__ void atomic_add_agent(float* p, float v) {
  __hip_atomic_fetch_add(p, v, __ATOMIC_RELAXED, __HIP_MEMORY_SCOPE_AGENT);
}

// Kernel 1: zero the accumulation output and build xs = x*scale - offset in ws.
__global__ __launch_bounds__(TPB) void ffd_prep(const float* __restrict__ x,
                                                const float* __restrict__ scale,
                                                const float* __restrict__ off,
                                                float* __restrict__ xs_g,
                                                float* __restrict__ out) {
  int t = blockIdx.x * TPB + threadIdx.x;
  if (t < OUT_ELEMS / 4) {
    f32x4v z = {0.f, 0.f, 0.f, 0.f};
    *(f32x4v*)(out + (size_t)t * 4) = z;
  }
  if (t < XS_ELEMS) {
    int comp = t % 3;
    xs_g[t] = x[t] * scale[comp] - off[comp];
  }
}

// Kernel 2: stage 96KB xs table into LDS via TDM (with a sequenced overwrite as
// a correctness hedge), then stream sorted COO triplets with b128 loads,
// run-length accumulate per thread, flush rows with agent-scope f32 atomics.
__global__ __launch_bounds__(TPB, 1) void ffd_spmm(const float* __restrict__ vals,
                                                   const int* __restrict__ rows,
                                                   const int* __restrict__ cols,
                                                   const float* __restrict__ xs_g,
                                                   float* __restrict__ out) {
  __shared__ float xs[XS_ELEMS];  // 96 KB

#if __has_builtin(__builtin_amdgcn_tensor_load_to_lds)
  // Tensor Data Mover: one 1-D descriptor copies the whole 96KB tile
  // global -> LDS. Issued by wave 0 only (EXEC is ignored by TDM, so other
  // waves must branch around it); tracked by TENSORcnt.
  if (threadIdx.x < 32) {
    uint32_t ldsOff = (uint32_t)(uintptr_t)&xs[0];           // low 32 bits = LDS byte offset
    uint64_t ga     = (uint64_t)(uintptr_t)xs_g;             // tile start (57-bit global addr)
    // D# group 0: count=1 valid, lds_addr, global_addr, type=2 ("image")
    u32x4v g0 = { 1u,
                  ldsOff,
                  (uint32_t)ga,
                  ((uint32_t)(ga >> 32) & 0x01FFFFFFu) | 0x80000000u };
    // D# group 1: data_size=4B (code 2); tensor_dim0 = tile_dim0 = 24576 elems;
    // tensor_dim1 = 1; dim0_stride = 24576; no padding/iterate/multicast/barrier.
    i32x8v g1 = { (int)(2u << 16),                              // [17:16] data_size=2
                  (int)(((unsigned)XS_ELEMS & 0xFFFFu) << 16),  // [63:48] tensor_dim0 lo16
                  (int)(1u << 16),                              // [95:80] tensor_dim1 = 1
                  (int)(((unsigned)XS_ELEMS & 0xFFFFu) << 16),  // [127:112] tile_dim0
                  0,                                            // tile_dim1/tile_dim2 = 0
                  (int)XS_ELEMS,                                // tensor_dim0_stride lo32
                  0, 0 };
    i32x4v gz4 = { 0, 0, 0, 0 };                                // groups 2/3 unused (<=2D)
    i32x8v gz8 = { 0, 0, 0, 0, 0, 0, 0, 0 };                    // extra group (6-arg form)
    __builtin_amdgcn_tensor_load_to_lds(g0, g1, gz4, gz4, gz8, 0);
#if __has_builtin(__builtin_amdgcn_s_wait_tensorcnt)
    __builtin_amdgcn_s_wait_tensorcnt(0);
#else
    asm volatile("s_wait_tensorcnt 0x0" ::: "memory");
#endif
    asm volatile("" ::: "memory");
  }
  __syncthreads();  // TDM writes (whatever they were) are complete device-side
#endif
  // Sequenced cooperative copy: if the (hardware-unverified) TDM descriptor did
  // the copy, this rewrites identical data; if not, it repairs the table.
  // Runs AFTER the tensorcnt wait + barrier so it cannot race the DMA.
  {
    const f32x4v* src = (const f32x4v*)xs_g;
    f32x4v*       dst = (f32x4v*)xs;
    for (int i = threadIdx.x; i < XS_ELEMS / 4; i += TPB) dst[i] = src[i];
  }
  __syncthreads();

  const int tid = threadIdx.x;
  for (int tile = blockIdx.x; tile < NTILES; tile += gridDim.x) {
    size_t base = (size_t)tile * TILE_NNZ + (size_t)tid * IPT;  // 16B-aligned per thread

    // Prefetch next grid-stride tile (global_prefetch_b8): one byte per lane,
    // lanes are 80B apart so the wave covers the whole next region's lines.
    if (tile + gridDim.x < NTILES) {
      size_t nk = base + (size_t)gridDim.x * TILE_NNZ;
      __builtin_prefetch(rows + nk, 0, 1);
      __builtin_prefetch(cols + nk, 0, 1);
      __builtin_prefetch(vals + nk, 0, 1);
    }

    int   cur = -1;
    float a0 = 0.f, a1 = 0.f, a2 = 0.f, a3 = 0.f, a4 = 0.f, a5 = 0.f;

#pragma unroll
    for (int g = 0; g < IPT / 4; ++g) {
      size_t k = base + (size_t)g * 4;
      i32x4v r4 = *(const i32x4v*)(rows + k);
      i32x4v c4 = *(const i32x4v*)(cols + k);
      f32x4v v4 = *(const f32x4v*)(vals + k);
#pragma unroll
      for (int j = 0; j < 4; ++j) {
        int   row = r4[j];
        int   col = c4[j];
        float v   = v4[j];
        if (row != cur) {
          if (cur >= 0) {
            float* o0 = out + (size_t)cur * 3;
            float* o1 = out + OUT_B1 + (size_t)cur * 3;
            atomic_add_agent(o0 + 0, a0);
            atomic_add_agent(o0 + 1, a1);
            atomic_add_agent(o0 + 2, a2);
            atomic_add_agent(o1 + 0, a3);
            atomic_add_agent(o1 + 1, a4);
            atomic_add_agent(o1 + 2, a5);
          }
          cur = row;
          a0 = a1 = a2 = a3 = a4 = a5 = 0.f;
        }
        int c0 = col * 3;
        a0 = fmaf(v, xs[c0 + 0], a0);
        a1 = fmaf(v, xs[c0 + 1], a1);
        a2 = fmaf(v, xs[c0 + 2], a2);
        a3 = fmaf(v, xs[XS_B1 + c0 + 0], a3);
        a4 = fmaf(v, xs[XS_B1 + c0 + 1], a4);
        a5 = fmaf(v, xs[XS_B1 + c0 + 2], a5);
      }
    }
    if (cur >= 0) {
      float* o0 = out + (size_t)cur * 3;
      float* o1 = out + OUT_B1 + (size_t)cur * 3;
      atomic_add_agent(o0 + 0, a0);
      atomic_add_agent(o0 + 1, a1);
      atomic_add_agent(o0 + 2, a2);
      atomic_add_agent(o1 + 0, a3);
      atomic_add_agent(o1 + 1, a4);
      atomic_add_agent(o1 + 2, a5);
    }
  }
}

extern "C" void kernel_launch(void* const* d_in, const int* in_sizes, int n_in,
                              void* d_out, int out_size, void* d_ws, size_t ws_size,
                              hipStream_t stream) {
  const float* x     = (const float*)d_in[0];  // [2, 4096, 3]
  const float* scale = (const float*)d_in[1];  // [3]
  const float* off   = (const float*)d_in[2];  // [3]
  const float* vals  = (const float*)d_in[3];  // [NNZ]
  const int*   rows  = (const int*)d_in[4];    // [NNZ] sorted
  const int*   cols  = (const int*)d_in[5];    // [NNZ]
  float*       out   = (float*)d_out;          // [2, 200000, 3]
  float*       xs_g  = (float*)d_ws;           // 96 KB scratch: scaled control points

  (void)in_sizes; (void)n_in; (void)out_size; (void)ws_size;

  int prep_blocks = (OUT_ELEMS / 4 + TPB - 1) / TPB;  // covers zeroing; xs is a subset
  ffd_prep<<<prep_blocks, TPB, 0, stream>>>(x, scale, off, xs_g, out);
  ffd_spmm<<<NBLOCKS, TPB, 0, stream>>>(vals, rows, cols, xs_g, out);
}